// OutlierSuppressionModule_49048526520863
// MI455X (gfx1250) — compile-verified
//
#include <hip/hip_runtime.h>
#include <hip/hip_bf16.h>
#include <stdint.h>

// Problem constants (fixed by the reference setup_inputs()).
#define B_  16
#define C_  1024
#define H_  32
#define W_  32
#define P_  1024    // H*W
#define NH_ 16
#define N_  1025    // 1 + H*W
#define K_  10      // TOP_K

typedef float v2f __attribute__((ext_vector_type(2)));
typedef float v8f __attribute__((ext_vector_type(8)));

__device__ __constant__ int c_dy[8] = {-1,-1,-1, 0, 0, 1, 1, 1};
__device__ __constant__ int c_dx[8] = {-1, 0, 1,-1, 1,-1, 0, 1};

__device__ __forceinline__ int clampi(int v, int lo, int hi) {
  return v < lo ? lo : (v > hi ? hi : v);
}

// ---------------------------------------------------------------------------
// 1) ratio[b][p] = mean_h attn[b,h,0,1+p] / (mean_h attn[b,h,1+p,1+p] + 1e-8)
// ---------------------------------------------------------------------------
__global__ void ratio_kernel(const float* __restrict__ attn, float* __restrict__ ratio) {
  int t = blockIdx.x * 256 + threadIdx.x;     // 0 .. B_*P_-1
  int b = t >> 10, p = t & 1023;
  size_t hb = (size_t)b * NH_ * (size_t)N_ * N_;
  float ss = 0.f, cs = 0.f;
  for (int h = 0; h < NH_; ++h) {
    size_t o = hb + (size_t)h * (size_t)N_ * N_;
    size_t d = (size_t)(1 + p);
    ss += attn[o + d * N_ + d];   // diagonal
    cs += attn[o + d];            // row 0
  }
  float sm = ss * (1.0f / 16.0f);
  float cm = cs * (1.0f / 16.0f);
  ratio[t] = cm / (sm + 1e-8f);
}

// ---------------------------------------------------------------------------
// 2) Per-batch top-10 (descending, ties -> lower index, like lax.top_k) and
//    "last writer" flags for the 80 neighbor slots (sequential fori_loop
//    semantics collapsed into collision-free parallel writes).
// ---------------------------------------------------------------------------
__global__ void topk_flags_kernel(const float* __restrict__ ratio,
                                  int* __restrict__ idxw, int* __restrict__ flags) {
  __shared__ float vals[1024];
  __shared__ float rv[512];
  __shared__ int   rid[512];
  __shared__ int   sidx[K_];
  __shared__ int   sy[80], sx[80], skp[80];
  int t = threadIdx.x, b = blockIdx.x;
  vals[t] = ratio[b * 1024 + t];
  __syncthreads();
  for (int r = 0; r < K_; ++r) {
    if (t < 512) {
      float v1 = vals[t];       int i1 = t;
      float v2 = vals[t + 512]; int i2 = t + 512;
      if (v2 > v1 || (v2 == v1 && i2 < i1)) { v1 = v2; i1 = i2; }
      rv[t] = v1; rid[t] = i1;
    }
    __syncthreads();
    for (int s = 256; s > 0; s >>= 1) {
      if (t < s) {
        float v2 = rv[t + s]; int i2 = rid[t + s];
        if (v2 > rv[t] || (v2 == rv[t] && i2 < rid[t])) { rv[t] = v2; rid[t] = i2; }
      }
      __syncthreads();
    }
    if (t == 0) { sidx[r] = rid[0]; vals[rid[0]] = -3.402823466e38f; }
    __syncthreads();
  }
  if (t < 80) {
    int i = t >> 3, j = t & 7;
    int id = sidx[i]; int rr = id >> 5, cc = id & 31;
    int y = clampi(rr + c_dy[j], 0, H_ - 1);
    int x = clampi(cc + c_dx[j], 0, W_ - 1);
    sy[t] = y; sx[t] = x;
    skp[t] = ((y != rr) || (x != cc)) ? 1 : 0;
  }
  __syncthreads();
  if (t < 80) {
    int f = skp[t];
    if (f) {
      for (int k2 = t + 1; k2 < 80; ++k2)
        if (skp[k2] && sy[k2] == sy[t] && sx[k2] == sx[t]) { f = 0; break; }
    }
    flags[b * 80 + t] = f;
  }
  if (t < K_) idxw[b * K_ + t] = sidx[t];
}

// ---------------------------------------------------------------------------
// 3) Bulk copy feature_map -> out through LDS with CDNA5 async transfers.
//    Each block moves 16 KB (1024 float4) via global_load_async_to_lds_b128 /
//    global_store_async_from_lds_b128, synced with s_wait_asynccnt.
// ---------------------------------------------------------------------------
__global__ void copy_async_kernel(const float4* __restrict__ src, float4* __restrict__ dst) {
  __shared__ float4 buf[1024];   // 16 KB
  int t = threadIdx.x;
  size_t base = (size_t)blockIdx.x * 1024;
  uint32_t lds0 = (uint32_t)(uintptr_t)(&buf[0]);
#pragma unroll
  for (int e = 0; e < 4; ++e) {
    size_t gi = base + (size_t)(e * 256 + t);
    uint32_t lds = lds0 + (uint32_t)((e * 256 + t) * 16);
    uint64_t gp = (uint64_t)(uintptr_t)(src + gi);
    asm volatile("global_load_async_to_lds_b128 %0, %1, off"
                 :: "v"(lds), "v"(gp) : "memory");
  }
  asm volatile("s_wait_asynccnt 0" ::: "memory");
#pragma unroll
  for (int e = 0; e < 4; ++e) {
    size_t gi = base + (size_t)(e * 256 + t);
    uint32_t lds = lds0 + (uint32_t)((e * 256 + t) * 16);
    uint64_t gp = (uint64_t)(uintptr_t)(dst + gi);
    asm volatile("global_store_async_from_lds_b128 %0, %1, off"
                 :: "v"(gp), "v"(lds) : "memory");
  }
  asm volatile("s_wait_asynccnt 0" ::: "memory");
}

// ---------------------------------------------------------------------------
// 4) Per (batch, outlier): Gram matrix G = V*V^T via V_WMMA_F32_16X16X4_F32,
//    where V rows are [center, nbr0..nbr7] (9x1024 f32, padded to 16 rows).
//    A (16x4) layout: lane l, vgpr v -> V[l%16][kb + 2*(l/16) + v].
//    B (4x16) layout is the transpose striping -> identical register content.
//    Inactive pad rows (9..15) read the (valid) center address and are zeroed
//    arithmetically (msk) so there is NO lane-divergent control flow: EXEC is
//    all-1s at every WMMA as the ISA requires. Two independent accumulator
//    chains per wave give load/WMMA overlap. 4 waves split K; deterministic
//    LDS combine (no float atomics).
// ---------------------------------------------------------------------------
__global__ void simw_kernel(const float* __restrict__ fm, const int* __restrict__ idxw,
                            float* __restrict__ wgt, float* __restrict__ strg) {
  __shared__ float g[1024];     // 4 waves x 256-entry D tiles
  int b = blockIdx.x / K_, i = blockIdx.x % K_;
  int id = idxw[b * K_ + i];
  int rr = id >> 5, cc = id & 31;
  int lane = threadIdx.x & 31;
  int wv   = threadIdx.x >> 5;
  int row  = lane & 15, half = lane >> 4;
  int y = rr, x = cc;                       // row 0 and pad rows: center pixel
  if (row >= 1 && row < 9) {
    y = clampi(rr + c_dy[row - 1], 0, H_ - 1);
    x = clampi(cc + c_dx[row - 1], 0, W_ - 1);
  }
  float msk = (row < 9) ? 1.0f : 0.0f;      // arithmetic zeroing of pad rows
  // Start at channel k = wv*4 + 2*half; channel stride is P_ floats.
  const float* p = fm + (size_t)b * ((size_t)C_ * P_) + (size_t)(y * W_ + x)
                      + (size_t)(wv * 4 + 2 * half) * P_;
  v8f acc0 = {}, acc1 = {};
  for (int t = 0; t < 32; ++t) {            // uniform trip count, no divergence
    v2f a0, a1;
    a0.x = p[0]             * msk;
    a0.y = p[(size_t)P_]    * msk;
    a1.x = p[(size_t)16 * P_] * msk;
    a1.y = p[(size_t)17 * P_] * msk;
    acc0 = __builtin_amdgcn_wmma_f32_16x16x4_f32(false, a0, false, a0,
                                                 (short)0, acc0, false, false);
    acc1 = __builtin_amdgcn_wmma_f32_16x16x4_f32(false, a1, false, a1,
                                                 (short)0, acc1, false, false);
    p += (size_t)32 * P_;
  }
  v8f acc = acc0 + acc1;
#pragma unroll
  for (int v = 0; v < 8; ++v) g[wv * 256 + v * 32 + lane] = acc[v];
  __syncthreads();
  for (int e = threadIdx.x; e < 256; e += 128)
    g[e] = g[e] + g[256 + e] + g[512 + e] + g[768 + e];
  __syncthreads();
  if (threadIdx.x == 0) {
    // G[M][N]: M<8 -> g[M*32+N]; M>=8 -> g[(M-8)*32+16+N]
    float n0 = fmaxf(sqrtf(g[0]), 1e-12f);          // ||center||
    float sim[8], z[8]; float zmax = -1e30f;
    for (int j = 0; j < 8; ++j) {
      int m = j + 1;
      float nsq = (m < 8) ? g[m * 32 + m] : g[24];  // G[8][8] -> g[0*32+16+8]
      float nn  = fmaxf(sqrtf(nsq), 1e-12f);
      sim[j] = g[m] / (nn * n0);                    // G[0][m]
      z[j]   = fmaxf(1.0f - sim[j], 0.0f);
      zmax   = fmaxf(zmax, z[j]);
    }
    float e8[8], es = 0.f;
    for (int j = 0; j < 8; ++j) { e8[j] = expf(z[j] - zmax); es += e8[j]; }
    float inv = 1.0f / es;
    int o = (b * K_ + i) * 8;
    for (int j = 0; j < 8; ++j) {
      wgt[o + j]  = e8[j] * inv;
      strg[o + j] = fminf(fmaxf(sim[j] * 0.1f, 0.0f), 1.0f);   // TEMP = 0.1
    }
  }
}

// ---------------------------------------------------------------------------
// 5) Neighbor "clean" writes (only last-writer slots): all reads from the
//    original fm (d_in), so fully parallel. L2-resident accesses.
// ---------------------------------------------------------------------------
__global__ void clean_kernel(const float* __restrict__ fm, float* __restrict__ out,
                             const int* __restrict__ idxw, const float* __restrict__ strg,
                             const int* __restrict__ flags) {
  int blk = blockIdx.x, b = blk / 80, k = blk % 80;
  if (flags[b * 80 + k] == 0) return;
  int i = k >> 3, j = k & 7;
  int id = idxw[b * K_ + i];
  int rr = id >> 5, cc = id & 31;
  int y = clampi(rr + c_dy[j], 0, H_ - 1);
  int x = clampi(cc + c_dx[j], 0, W_ - 1);
  float s = strg[b * 80 + k];
  size_t pb = (size_t)b * ((size_t)C_ * P_);
  int pos = y * W_ + x, cp = rr * W_ + cc;
#pragma unroll
  for (int e = 0; e < 4; ++e) {
    int c = threadIdx.x + e * 256;
    size_t off = pb + (size_t)c * P_;
    out[off + pos] = fm[off + pos] - fm[off + cp] * s;
  }
}

// ---------------------------------------------------------------------------
// 6) Final weighted-average writes at the outlier centers (overrides all).
// ---------------------------------------------------------------------------
__global__ void wavg_kernel(const float* __restrict__ fm, float* __restrict__ out,
                            const int* __restrict__ idxw, const float* __restrict__ wgt) {
  int b = blockIdx.x / K_, i = blockIdx.x % K_;
  int id = idxw[b * K_ + i];
  int rr = id >> 5, cc = id & 31;
  float wj[8]; int pj[8];
#pragma unroll
  for (int j = 0; j < 8; ++j) {
    int y = clampi(rr + c_dy[j], 0, H_ - 1);
    int x = clampi(cc + c_dx[j], 0, W_ - 1);
    pj[j] = y * W_ + x;
    wj[j] = wgt[(b * K_ + i) * 8 + j];
  }
  size_t pb = (size_t)b * ((size_t)C_ * P_);
  int cp = rr * W_ + cc;
#pragma unroll
  for (int e = 0; e < 4; ++e) {
    int c = threadIdx.x + e * 256;
    size_t off = pb + (size_t)c * P_;
    float a = 0.f;
#pragma unroll
    for (int j = 0; j < 8; ++j) a += wj[j] * fm[off + pj[j]];
    out[off + cp] = a;
  }
}

// ---------------------------------------------------------------------------
extern "C" void kernel_launch(void* const* d_in, const int* in_sizes, int n_in,
                              void* d_out, int out_size, void* d_ws, size_t ws_size,
                              hipStream_t stream) {
  (void)in_sizes; (void)n_in; (void)out_size; (void)ws_size;
  const float* fm   = (const float*)d_in[0];
  const float* attn = (const float*)d_in[1];
  float* out = (float*)d_out;

  char*  w     = (char*)d_ws;
  float* ratio = (float*)(w);              // B_*P_ floats        = 65536 B
  int*   idxw  = (int*)  (w + 65536);      // B_*K_ ints          =   640 B
  float* wgt   = (float*)(w + 66176);      // B_*80 floats        =  5120 B
  float* strg  = (float*)(w + 71296);      // B_*80 floats        =  5120 B
  int*   flags = (int*)  (w + 76416);      // B_*80 ints          =  5120 B

  ratio_kernel     <<<(B_ * P_) / 256, 256, 0, stream>>>(attn, ratio);
  topk_flags_kernel<<<B_, 1024, 0, stream>>>(ratio, idxw, flags);
  // Copy before simw so the feature map is L2-resident for the scattered reads.
  copy_async_kernel<<<(B_ * C_ * P_) / (4 * 1024), 256, 0, stream>>>(
      (const float4*)fm, (float4*)out);
  simw_kernel      <<<B_ * K_, 128, 0, stream>>>(fm, idxw, wgt, strg);
  clean_kernel     <<<B_ * 80, 256, 0, stream>>>(fm, out, idxw, strg, flags);
  wavg_kernel      <<<B_ * K_, 256, 0, stream>>>(fm, out, idxw, wgt);
}